// ModelNew_73315091744130
// MI455X (gfx1250) — compile-verified
//
#include <hip/hip_runtime.h>

typedef float v2f __attribute__((ext_vector_type(2)));
typedef float v8f __attribute__((ext_vector_type(8)));
typedef unsigned int u32x4 __attribute__((ext_vector_type(4)));
typedef int i32x4 __attribute__((ext_vector_type(4)));
typedef int i32x8 __attribute__((ext_vector_type(8)));

#define IC     64
#define OC     128
#define KSIZE  576          // 64*3*3, K index = ic*9 + kh*3 + kw (weight native layout)
#define H      256
#define W      256
#define OHH    254
#define OWW    254
#define COLS   128          // output columns per block
#define TILES  8            // 16-col WMMA tiles per wave
#define WLDS   132          // 128 cols + 2 halo + 2 pad (multiple of 4B rows for TDM)
#define XLDSF  (IC * 3 * WLDS)   // 25344 floats staged x slab
#define NWAVES 8

#if __has_builtin(__builtin_amdgcn_tensor_load_to_lds) && __has_builtin(__builtin_amdgcn_s_wait_tensorcnt)
#define USE_TDM 1
#else
#define USE_TDM 0
#endif

__global__ __launch_bounds__(256)
void conv3x3_bias_scale_minC_wmma(const float* __restrict__ x,
                                  const float* __restrict__ wgt,
                                  const float* __restrict__ bias,
                                  float* __restrict__ out)
{
    extern __shared__ float lds[];  // [XLDSF] x slab | [NWAVES*COLS] reduction
    const int tid  = threadIdx.x;
    const int lane = tid & 31;
    const int wave = tid >> 5;          // 0..7 -> oc group of 16
    const int l15  = lane & 15;
    const int lhi  = lane >> 4;         // 0 or 1 (K/M half select)

    const int bx   = blockIdx.x;
    const int g    = bx & 1;            // column group
    const int row  = bx >> 1;           // n*254 + oh
    const int n    = row / OHH;
    const int oh   = row - n * OHH;
    const int col0 = g * COLS;

    const float* xin = x + (size_t)n * IC * H * W;

#if USE_TDM
    // ---- TDM: one 3D descriptor moves x[n, 0:64, oh:oh+3, col0:col0+132] -> LDS ----
    if (tid < 32) {
        unsigned long long ga = (unsigned long long)(uintptr_t)(xin + (size_t)oh * W + col0);
        unsigned int ldsAddr  = (unsigned int)(uintptr_t)(&lds[0]);   // low 32 bits = LDS offset
        unsigned int td0      = (unsigned int)(W - col0);             // OOB cols -> 0 (halo)

        u32x4 g0;
        g0[0] = 1u;                                     // count=1, user descriptor
        g0[1] = ldsAddr;                                // lds_addr
        g0[2] = (unsigned int)ga;                       // global_addr[31:0]
        g0[3] = ((unsigned int)(ga >> 32) & 0x1FFFFFFu) // global_addr[56:32]
                | (2u << 30);                           // type = 2 ("image")

        i32x8 g1;
        g1[0] = (int)(2u << 16);                        // data_size = 2 (4 bytes)
        g1[1] = (int)((td0 & 0xFFFFu) << 16);           // tensor_dim0[15:0]
        g1[2] = (int)((td0 >> 16) | (3u << 16));        // tensor_dim0[31:16] | tensor_dim1=3
        g1[3] = (int)(0u | ((unsigned)WLDS << 16));     // tensor_dim1[31:16]=0 | tile_dim0=132
        g1[4] = (int)(3u | (64u << 16));                // tile_dim1=3 | tile_dim2=64
        g1[5] = (int)(unsigned)W;                       // tensor_dim0_stride[31:0] = 256
        g1[6] = (int)(0u | ((unsigned)(H * W) & 0xFFFFu) << 16);   // | dim1_stride[15:0]
        g1[7] = (int)((unsigned)(H * W) >> 16);         // tensor_dim1_stride[47:16]

        i32x4 g2;
        g2[0] = IC;                                     // tensor_dim2 = 64
        g2[1] = 0; g2[2] = 0; g2[3] = 0;                // no dim3 / no iterate
        i32x4 g3; g3[0] = 0; g3[1] = 0; g3[2] = 0; g3[3] = 0;

#if __has_include(<hip/amd_detail/amd_gfx1250_TDM.h>)
        i32x8 g4; g4[0]=0; g4[1]=0; g4[2]=0; g4[3]=0; g4[4]=0; g4[5]=0; g4[6]=0; g4[7]=0;
        __builtin_amdgcn_tensor_load_to_lds(g0, g1, g2, g3, g4, 0);  // clang-23 6-arg form
#else
        __builtin_amdgcn_tensor_load_to_lds(g0, g1, g2, g3, 0);      // ROCm 7.2 5-arg form
#endif
        __builtin_amdgcn_s_wait_tensorcnt(0);
    }
#else
    // ---- fallback: cooperative staging (clamped at W-1) ----
    for (int idx = tid; idx < XLDSF; idx += 256) {
        int ic  = idx / (3 * WLDS);
        int rem = idx - ic * (3 * WLDS);
        int h   = rem / WLDS;
        int j   = rem - h * WLDS;
        int gw  = col0 + j; if (gw > W - 1) gw = W - 1;
        lds[idx] = xin[(size_t)ic * (H * W) + (oh + h) * W + gw];
    }
#endif
    __syncthreads();

    // ---- implicit GEMM: (16 oc per wave) x (128 pixels), K = 576, fp32 WMMA ----
    const int oc0 = wave * 16;
    // A 16x4 fragment: M = lane%16, K offset = 2*(lane>>4)  (ISA 7.12.2 layout)
    const float* awp = wgt + (size_t)(oc0 + l15) * KSIZE + (lhi << 1);
    const int klane = lhi << 1;

    v8f acc[TILES];
    v8f zero = {};
    #pragma unroll
    for (int t = 0; t < TILES; ++t) acc[t] = zero;

    v2f a_cur = *(const v2f*)awp;   // k0 = 0

    #pragma unroll 1
    for (int k0 = 0; k0 < KSIZE; k0 += 4) {
        int kn = k0 + 4; if (kn > KSIZE - 4) kn = KSIZE - 4;
        v2f a_nxt = *(const v2f*)(awp + kn);        // double-buffered A fragment

        // decompose the two per-lane K indices into (ic, kh, kw) -> LDS row bases
        int kk0 = k0 + klane;
        int ic0 = kk0 / 9; int r0 = kk0 - 9 * ic0; int kh0 = r0 / 3; int kw0 = r0 - 3 * kh0;
        int kk1 = kk0 + 1;
        int ic1 = kk1 / 9; int r1 = kk1 - 9 * ic1; int kh1 = r1 / 3; int kw1 = r1 - 3 * kh1;
        const float* b0p = lds + (ic0 * 3 + kh0) * WLDS + kw0 + l15;
        const float* b1p = lds + (ic1 * 3 + kh1) * WLDS + kw1 + l15;

        #pragma unroll
        for (int t = 0; t < TILES; ++t) {
            v2f b;
            b.x = b0p[t * 16];   // B[k0+klane  ][16t + lane%16]
            b.y = b1p[t * 16];   // B[k0+klane+1][16t + lane%16]
            acc[t] = __builtin_amdgcn_wmma_f32_16x16x4_f32(
                         false, a_cur, false, b, (short)0, acc[t], false, false);
        }
        a_cur = a_nxt;
    }

    // ---- fused epilogue: (y + bias) * 0.5, min over 128 channels ----
    float bb[8];
    #pragma unroll
    for (int r = 0; r < 8; ++r) bb[r] = bias[oc0 + (lhi << 3) + r];  // M = r + 8*lhi

    float* red = lds + XLDSF + wave * COLS;
    #pragma unroll
    for (int t = 0; t < TILES; ++t) {
        float m = (acc[t][0] + bb[0]) * 0.5f;
        #pragma unroll
        for (int r = 1; r < 8; ++r)
            m = fminf(m, (acc[t][r] + bb[r]) * 0.5f);
        // merge the two M-halves (lane L <-> L^16 hold same column, other 8 ocs)
        m = fminf(m, __shfl_xor(m, 16, 32));
        if (lhi == 0) red[t * 16 + l15] = m;     // per-wave min of its 16 ocs
    }
    __syncthreads();

    // cross-wave min over the 8 oc-groups, then store the row segment
    if (tid < COLS) {
        int col = col0 + tid;
        if (col < OWW) {
            float m = lds[XLDSF + tid];
            #pragma unroll
            for (int w2 = 1; w2 < NWAVES; ++w2)
                m = fminf(m, lds[XLDSF + w2 * COLS + tid]);
            out[((size_t)n * OHH + oh) * OWW + col] = m;
        }
    }
}

extern "C" void kernel_launch(void* const* d_in, const int* in_sizes, int n_in,
                              void* d_out, int out_size, void* d_ws, size_t ws_size,
                              hipStream_t stream) {
    const float* x    = (const float*)d_in[0];
    const float* wgt  = (const float*)d_in[1];
    const float* bias = (const float*)d_in[2];
    float* out        = (float*)d_out;

    const int grid = 16 * OHH * 2;                        // (n, oh, col-group)
    const size_t shbytes = (XLDSF + NWAVES * COLS) * sizeof(float);  // ~105 KB
    conv3x3_bias_scale_minC_wmma<<<grid, 256, shbytes, stream>>>(x, wgt, bias, out);
}